// LSTMNet_62577673503142
// MI455X (gfx1250) — compile-verified
//
#include <hip/hip_runtime.h>
#include <hip/hip_bf16.h>

typedef float v2f __attribute__((ext_vector_type(2)));
typedef float v8f __attribute__((ext_vector_type(8)));

#define Bn 512
#define Tn 512
#define En 100
#define Hn 100
#define G4 400   // 4*H
#define KP 50    // K pairs (100/2)
#define NT 25    // N tiles (400/16)
#define KC 25    // K chunks of 4 (100/4)

// ---------------- Prep: pack W_ih^T and W_hh^T as float2 K-pairs, fuse biases ----
// WihP[kk*400 + n] = (W_ih[n][2kk], W_ih[n][2kk+1]);  same for WhhP.  bias[g]=b_ih[g]+b_hh[g]
__global__ __launch_bounds__(256) void prep_kernel(
    const float* __restrict__ W_ih, const float* __restrict__ W_hh,
    const float* __restrict__ b_ih, const float* __restrict__ b_hh,
    v2f* __restrict__ WihP, v2f* __restrict__ WhhP, float* __restrict__ bias)
{
    int idx = blockIdx.x * 256 + threadIdx.x;
    if (idx < 20000) {
        int kk = idx / G4, n = idx % G4;
        v2f v; v.x = W_ih[n * En + 2 * kk]; v.y = W_ih[n * En + 2 * kk + 1];
        WihP[idx] = v;
    } else if (idx < 40000) {
        int j = idx - 20000;
        int kk = j / G4, n = j % G4;
        v2f v; v.x = W_hh[n * Hn + 2 * kk]; v.y = W_hh[n * Hn + 2 * kk + 1];
        WhhP[j] = v;
    } else if (idx < 40400) {
        int g = idx - 40000;
        bias[g] = b_ih[g] + b_hh[g];
    }
}

// ---------------- Phase 1: xg[t*B+b][g] = emb[x[b][t]] . W_ih[g] + bias[g] -------
// Block: 128 rows (8 M-tiles) x 25 N-tiles, 256 threads = 8 wave32s.
__global__ __launch_bounds__(256) void embed_gemm_kernel(
    const int* __restrict__ x, const float* __restrict__ emb,
    const v2f* __restrict__ WihP, const float* __restrict__ bias,
    float* __restrict__ xg)
{
    __shared__ int s_tok[128];
    __shared__ v2f s_A[128 * KP];   // 51.2 KB, packed embedding rows

    const int tid  = threadIdx.x;
    const int lane = tid & 31;
    const int wave = tid >> 5;
    const int half = lane >> 4;
    const int lx   = lane & 15;
    const long base = (long)blockIdx.x * 128;   // flattened row = t*B + b

    if (tid < 128) {
        long row = base + tid;
        int t = (int)(row >> 9);       // /512
        int b = (int)(row & 511);
        s_tok[tid] = x[(b << 9) + t];  // x is [B][T], T=512
    }
    __syncthreads();

    for (int idx = tid; idx < 128 * KP; idx += 256) {
        int rr = idx / KP, kk = idx % KP;
        s_A[idx] = ((const v2f*)emb)[(long)s_tok[rr] * KP + kk];
    }
    __syncthreads();

    for (int tile = wave; tile < 8 * NT; tile += 8) {
        int mt = tile / NT, nt = tile % NT;
        int n0 = nt * 16;
        float bz = bias[n0 + lx];
        v8f d;
#pragma unroll
        for (int r = 0; r < 8; ++r) d[r] = bz;
#pragma unroll
        for (int kc = 0; kc < KC; ++kc) {
            v2f a  = s_A[(mt * 16 + lx) * KP + 2 * kc + half];
            v2f bb = WihP[(2 * kc + half) * G4 + n0 + lx];
            d = __builtin_amdgcn_wmma_f32_16x16x4_f32(false, a, false, bb,
                                                      (short)0, d, false, false);
        }
        long rowg = base + mt * 16;
#pragma unroll
        for (int r = 0; r < 8; ++r)
            xg[(rowg + r + 8 * half) * G4 + n0 + lx] = d[r];
    }
}

// ---------------- Phase 2: recurrent LSTM over T, 16 batch rows per block -------
// Dynamic LDS: WhhP(160000) | gates(25600) | h(6400) | c(6400) | pool(6400) = 204800 B
__global__ __launch_bounds__(256) void lstm_kernel(
    const float* __restrict__ xg, const v2f* __restrict__ WhhP,
    const float* __restrict__ fc_w, const float* __restrict__ fc_b,
    float* __restrict__ out)
{
    extern __shared__ char smem[];
    v2f*   sWhh   = (v2f*)smem;                       // [50*400] packed K-pairs
    float* sGates = (float*)(smem + 160000);          // [16][400]
    float* sH     = (float*)(smem + 185600);          // [16][100] (v2f-aligned)
    float* sC     = (float*)(smem + 192000);          // [16][100]
    float* sPool  = (float*)(smem + 198400);          // [16][100]

    const int tid  = threadIdx.x;
    const int lane = tid & 31;
    const int wave = tid >> 5;
    const int half = lane >> 4;
    const int lx   = lane & 15;
    const int b0   = blockIdx.x * 16;

    for (int idx = tid; idx < KP * G4; idx += 256) sWhh[idx] = WhhP[idx];
    for (int idx = tid; idx < 16 * Hn; idx += 256) {
        sH[idx] = 0.0f; sC[idx] = 0.0f; sPool[idx] = 0.0f;
    }
    __syncthreads();

    for (int t = 0; t < Tn; ++t) {
        // gates = xg[t, b0:b0+16, :] + h @ W_hh^T
        const float* xgp = xg + ((long)t * Bn + b0) * G4;
        for (int nt = wave; nt < NT; nt += 8) {
            int n0 = nt * 16;
            v8f d;
#pragma unroll
            for (int r = 0; r < 8; ++r)
                d[r] = xgp[(long)(r + 8 * half) * G4 + n0 + lx];
#pragma unroll
            for (int kc = 0; kc < KC; ++kc) {
                v2f a  = ((const v2f*)sH)[lx * KP + 2 * kc + half];
                v2f bb = sWhh[(2 * kc + half) * G4 + n0 + lx];
                d = __builtin_amdgcn_wmma_f32_16x16x4_f32(false, a, false, bb,
                                                          (short)0, d, false, false);
            }
#pragma unroll
            for (int r = 0; r < 8; ++r)
                sGates[(r + 8 * half) * G4 + n0 + lx] = d[r];
        }
        __syncthreads();

        // elementwise gates (PyTorch order i,f,g,o), update c,h, accumulate pool
        for (int idx = tid; idx < 16 * Hn; idx += 256) {
            int b = idx / Hn, hh = idx % Hn;
            const float* gr = sGates + b * G4 + hh;
            float gi = gr[0], gf = gr[Hn], gg = gr[2 * Hn], go = gr[3 * Hn];
            float si = 1.0f / (1.0f + __expf(-gi));
            float sf = 1.0f / (1.0f + __expf(-gf));
            float so = 1.0f / (1.0f + __expf(-go));
            float c  = sf * sC[idx] + si * tanhf(gg);
            float h  = so * tanhf(c);
            sC[idx] = c;
            sH[idx] = h;
            sPool[idx] += h;
        }
        __syncthreads();
    }

    for (int idx = tid; idx < 16 * Hn; idx += 256)
        sPool[idx] *= (1.0f / (float)Tn);
    __syncthreads();

    // FC: out[b][j] = pool[b] . fc_w[j] + fc_b[j]
    if (tid < 32) {
        int b = tid >> 1, j = tid & 1;
        float s = fc_b[j];
        for (int hh = 0; hh < Hn; ++hh)
            s += sPool[b * Hn + hh] * fc_w[j * Hn + hh];
        out[(b0 + b) * 2 + j] = s;
    }
}

extern "C" void kernel_launch(void* const* d_in, const int* in_sizes, int n_in,
                              void* d_out, int out_size, void* d_ws, size_t ws_size,
                              hipStream_t stream) {
    const int*   x    = (const int*)d_in[0];
    const float* emb  = (const float*)d_in[1];
    const float* W_ih = (const float*)d_in[2];
    const float* W_hh = (const float*)d_in[3];
    const float* b_ih = (const float*)d_in[4];
    const float* b_hh = (const float*)d_in[5];
    const float* fc_w = (const float*)d_in[6];
    const float* fc_b = (const float*)d_in[7];
    float* out = (float*)d_out;

    // workspace layout
    char* ws = (char*)d_ws;
    float* xg   = (float*)ws;                                   // 512*512*400*4 = 419,430,400 B
    size_t off  = (size_t)Bn * Tn * G4 * sizeof(float);
    v2f*   WihP = (v2f*)(ws + off);   off += 160000;
    v2f*   WhhP = (v2f*)(ws + off);   off += 160000;
    float* bias = (float*)(ws + off);

    prep_kernel<<<(40400 + 255) / 256, 256, 0, stream>>>(W_ih, W_hh, b_ih, b_hh,
                                                         WihP, WhhP, bias);

    embed_gemm_kernel<<<(Bn * Tn) / 128, 256, 0, stream>>>(x, emb, WihP, bias, xg);

    lstm_kernel<<<Bn / 16, 256, 204800, stream>>>(xg, WhhP, fc_w, fc_b, out);

    (void)in_sizes; (void)n_in; (void)out_size; (void)ws_size;
}